// Net6_81595788689559
// MI455X (gfx1250) — compile-verified
//
#include <hip/hip_runtime.h>

typedef float v2f __attribute__((ext_vector_type(2)));
typedef float v8f __attribute__((ext_vector_type(8)));

#define TB 256

// ---------------- elementwise / setup kernels ----------------

__global__ void k_fill(float* p, float v, int n) {
    int i = blockIdx.x * blockDim.x + threadIdx.x;
    if (i < n) p[i] = v;
}

__global__ void k_deg(const int* __restrict__ dst, int e, float* deg) {
    int i = blockIdx.x * blockDim.x + threadIdx.x;
    if (i < e) atomicAdd(&deg[dst[i]], 1.0f);
}

__global__ void k_rsqrt(float* d, int n) {
    int i = blockIdx.x * blockDim.x + threadIdx.x;
    if (i < n) d[i] = rsqrtf(d[i]);   // deg >= 1 always (self loop)
}

__global__ void k_norm(const int* __restrict__ src, const int* __restrict__ dst,
                       const float* __restrict__ dinv, float* __restrict__ nrm, int e) {
    int i = blockIdx.x * blockDim.x + threadIdx.x;
    if (i < e) nrm[i] = dinv[src[i]] * dinv[dst[i]];
}

// Pack weight [Kact x Dact] row-major into WMMA B-fragment-major layout:
// Wf[kb][j][lane][p] = W[kb*4 + (lane>>4)*2 + p][j*16 + (lane&15)]  (0 if OOB)
// so each lane's B fragment for (k-step kb, n-tile j) is one contiguous v2f.
__global__ void k_packfrag(const float* __restrict__ w, int Kact, int Dact,
                           float* __restrict__ wf, int Kp, int NT) {
    int i = blockIdx.x * blockDim.x + threadIdx.x;
    int total = (Kp / 4) * NT * 64;
    if (i >= total) return;
    int p    = i & 1;
    int lane = (i >> 1) & 31;
    int j    = (i >> 6) % NT;
    int kb   = i / (64 * NT);
    int row  = kb * 4 + (lane >> 4) * 2 + p;
    int col  = j * 16 + (lane & 15);
    wf[i] = (row < Kact && col < Dact) ? w[row * Dact + col] : 0.0f;
}

// ---------------- WMMA fp32 GEMM ----------------
// C[N x ldc] = A[N x K] * W[K x (NT*16)] (+bias broadcast over rows).
// One wave owns a 16-row strip and ALL NT column tiles: per K-step the A
// fragment is loaded once (b64) and NT pre-packed B fragments (one b64 each)
// feed NT independent WMMAs (distinct accumulators -> no RAW hazards).
template <int NT>
__global__ void k_gemm_wmma(const float* __restrict__ A, int lda, int K,
                            const float* __restrict__ Wf,
                            const float* __restrict__ bias, int dact,
                            float* __restrict__ C, int ldc, int mtiles) {
    int wave = (blockIdx.x * blockDim.x + threadIdx.x) >> 5;
    int lane = threadIdx.x & 31;
    if (wave >= mtiles) return;                // wave-uniform: EXEC all-1 inside
    int m0 = wave * 16;
    int half = lane >> 4;                      // 0: lanes 0-15, 1: lanes 16-31
    int l = lane & 15;

    // C/D layout: VGPR v -> row m0 + half*8 + v, col j*16 + l
    v8f acc[NT];
#pragma unroll
    for (int j = 0; j < NT; ++j) {
        float cinit = 0.0f;
        int col = j * 16 + l;
        if (bias != nullptr && col < dact) cinit = bias[col];
#pragma unroll
        for (int v = 0; v < 8; ++v) acc[j][v] = cinit;
    }

    // A 16x4 frag: lane l (half 0) holds A[m0+l][k..k+1]; half 1 holds k+2..k+3
    const float* arow = A + (size_t)(m0 + l) * lda + half * 2;   // 8B aligned (lda even)
    const v2f* bfrag = (const v2f*)Wf + lane;                    // fragment-major

    const int KB = K / 4;
    for (int kb = 0; kb < KB; ++kb) {
        v2f a = *(const v2f*)(arow + kb * 4);
        v2f b[NT];
#pragma unroll
        for (int j = 0; j < NT; ++j)
            b[j] = bfrag[(size_t)(kb * NT + j) * 32];
#pragma unroll
        for (int j = 0; j < NT; ++j)
            acc[j] = __builtin_amdgcn_wmma_f32_16x16x4_f32(false, a, false, b[j],
                                                           (short)0, acc[j], false, false);
    }

    float* crow = C + (size_t)(m0 + half * 8) * ldc + l;
#pragma unroll
    for (int j = 0; j < NT; ++j)
#pragma unroll
        for (int v = 0; v < 8; ++v) crow[(size_t)v * ldc + j * 16] = acc[j][v];
}

// ---------------- GCN aggregation ----------------

// out[i,f] = bias[f] + dinv[i]^2 * h2[i,f]  (self loop),  pad cols -> 0
__global__ void k_agg_init(const float* __restrict__ h2, const float* __restrict__ dinv,
                           const float* __restrict__ bias, int D, int Dp,
                           float* __restrict__ out, int n) {
    int i = blockIdx.x * blockDim.x + threadIdx.x;
    if (i >= n * Dp) return;
    int node = i / Dp, f = i % Dp;
    float v = 0.0f;
    if (f < D) {
        float di = dinv[node];
        v = bias[f] + di * di * h2[i];
    }
    out[i] = v;
}

// one WAVE per edge: edge index is wave-uniform so src/dst/norm fetches
// scalarize (SMEM), lanes stride the feature dim with coalesced loads +
// global_atomic_add_f32.
__global__ void k_scatter(const int* __restrict__ src, const int* __restrict__ dst,
                          const float* __restrict__ nrm, const float* __restrict__ h2,
                          float* __restrict__ out, int E, int D, int Dp) {
    int wave = (blockIdx.x * blockDim.x + threadIdx.x) >> 5;
    int lane = threadIdx.x & 31;
    if (wave >= E) return;
    int s = src[wave];
    int d = dst[wave];
    float w = nrm[wave];
    const float* hs = h2 + (size_t)s * Dp;
    float* od = out + (size_t)d * Dp;
    for (int f = lane; f < D; f += 32)
        atomicAdd(&od[f], w * hs[f]);
}

__global__ void k_relu(float* p, int n) {
    int i = blockIdx.x * blockDim.x + threadIdx.x;
    if (i < n) p[i] = fmaxf(p[i], 0.0f);
}

// ---------------- BatchNorm (batch statistics) ----------------

__global__ void k_bn_stats(const float* __restrict__ h, int n, int ld, int D,
                           int rowsPerBlock, float* sum, float* sumsq) {
    int f = threadIdx.x;
    if (f >= D) return;
    int r0 = blockIdx.x * rowsPerBlock;
    int r1 = min(r0 + rowsPerBlock, n);
    float s = 0.0f, sq = 0.0f;
    for (int r = r0; r < r1; ++r) {
        float v = h[(size_t)r * ld + f];
        s += v; sq += v * v;
    }
    atomicAdd(&sum[f], s);
    atomicAdd(&sumsq[f], sq);
}

__global__ void k_bn_apply(const float* __restrict__ in, int ldin,
                           float* __restrict__ out, int ldout,
                           int n, int D,
                           const float* __restrict__ sum, const float* __restrict__ sumsq,
                           const float* __restrict__ g, const float* __restrict__ b,
                           float inv_n, int do_relu) {
    int i = blockIdx.x * blockDim.x + threadIdx.x;
    if (i >= n * ldout) return;
    int node = i / ldout, f = i % ldout;
    float v = 0.0f;
    if (f < D) {
        float mu = sum[f] * inv_n;
        float var = sumsq[f] * inv_n - mu * mu;
        v = (in[(size_t)node * ldin + f] - mu) * rsqrtf(var + 1e-5f) * g[f] + b[f];
        if (do_relu) v = fmaxf(v, 0.0f);
    }
    out[i] = v;
}

// ---------------- final projection (D3 -> 1) ----------------

__global__ void k_fc3(const float* __restrict__ h, int ld, int D,
                      const float* __restrict__ w, const float* __restrict__ b,
                      float* __restrict__ out, int n) {
    int i = blockIdx.x * blockDim.x + threadIdx.x;
    if (i >= n) return;
    float s = b[0];
#pragma unroll 10
    for (int f = 0; f < D; ++f) s += h[(size_t)i * ld + f] * w[f];
    out[i] = s;
}

// ---------------- host orchestration ----------------

extern "C" void kernel_launch(void* const* d_in, const int* in_sizes, int n_in,
                              void* d_out, int out_size, void* d_ws, size_t ws_size,
                              hipStream_t stream) {
    const float* x       = (const float*)d_in[0];
    const int*   ei      = (const int*)  d_in[1];   // [2, E]
    const float* conv1_w = (const float*)d_in[2];
    const float* conv1_b = (const float*)d_in[3];
    const float* convs_w = (const float*)d_in[4];
    const float* convs_b = (const float*)d_in[5];
    const float* bn1_g   = (const float*)d_in[6];
    const float* bn1_b   = (const float*)d_in[7];
    const float* fc1_w   = (const float*)d_in[8];
    const float* fc1_b   = (const float*)d_in[9];
    const float* bn2_g   = (const float*)d_in[10];
    const float* bn2_b   = (const float*)d_in[11];
    const float* fc2_w   = (const float*)d_in[12];
    const float* fc2_b   = (const float*)d_in[13];
    const float* bn3_g   = (const float*)d_in[14];
    const float* bn3_b   = (const float*)d_in[15];
    const float* fc3_w   = (const float*)d_in[16];
    const float* fc3_b   = (const float*)d_in[17];
    float* out = (float*)d_out;

    const int FIN = 128, D1 = 90, D2 = 80, D3 = 50;
    const int P1 = 96, P2 = 80, P3 = 64;          // 16-padded dims
    const int N = in_sizes[0] / FIN;              // 50000 (multiple of 16)
    const int E = in_sizes[1] / 2;                // 800000
    const int* esrc = ei;
    const int* edst = ei + E;

    // workspace carve-up
    float* ws = (float*)d_ws;
    size_t off = 0;
    auto alloc = [&](size_t cnt) { float* p = ws + off; off += (cnt + 15) & ~(size_t)15; return p; };
    float* dinv = alloc(N);
    float* nrm  = alloc(E);
    float* hA   = alloc((size_t)N * P1);
    float* hB   = alloc((size_t)N * P1);
    float* h2   = alloc((size_t)N * P1);
    float* Wp   = alloc((size_t)FIN * P1);        // fragment-packed weights
    float* bsum = alloc(P1);
    float* bsq  = alloc(P1);
    (void)ws_size; (void)n_in; (void)out_size;

    const int mt = N / 16;                        // 3125 row-strip waves per GEMM

    // ---- normalization: deg (with self loop) -> dinv -> edge norms ----
    k_fill  <<<(N + TB - 1) / TB, TB, 0, stream>>>(dinv, 1.0f, N);
    k_deg   <<<(E + TB - 1) / TB, TB, 0, stream>>>(edst, E, dinv);
    k_rsqrt <<<(N + TB - 1) / TB, TB, 0, stream>>>(dinv, N);
    k_norm  <<<(E + TB - 1) / TB, TB, 0, stream>>>(esrc, edst, dinv, nrm, E);

    // pack W[Kact x Dact] -> fragment layout for Kp x (nt*16), then GEMM
    auto gemm = [&](const float* A, int lda, int Kp, const float* w, int Kact, int Dact,
                    const float* bias, float* C, int ldc, int nt) {
        int tot = (Kp / 4) * nt * 64;
        k_packfrag<<<(tot + TB - 1) / TB, TB, 0, stream>>>(w, Kact, Dact, Wp, Kp, nt);
        int blocks = (mt * 32 + TB - 1) / TB;
        if (nt == 6)
            k_gemm_wmma<6><<<blocks, TB, 0, stream>>>(A, lda, Kp, Wp, bias, Dact, C, ldc, mt);
        else if (nt == 5)
            k_gemm_wmma<5><<<blocks, TB, 0, stream>>>(A, lda, Kp, Wp, bias, Dact, C, ldc, mt);
        else
            k_gemm_wmma<4><<<blocks, TB, 0, stream>>>(A, lda, Kp, Wp, bias, Dact, C, ldc, mt);
    };

    auto conv = [&](const float* hin, int lda, int Kp, const float* w, int Kact,
                    const float* bias, float* hout) {
        gemm(hin, lda, Kp, w, Kact, D1, nullptr, h2, P1, P1 / 16);
        k_agg_init<<<((size_t)N * P1 + TB - 1) / TB, TB, 0, stream>>>(h2, dinv, bias, D1, P1, hout, N);
        k_scatter<<<((size_t)E * 32 + TB - 1) / TB, TB, 0, stream>>>(esrc, edst, nrm, h2, hout, E, D1, P1);
        k_relu<<<((size_t)N * P1 + TB - 1) / TB, TB, 0, stream>>>(hout, N * P1);
    };

    auto bn = [&](const float* in, int ldin, float* o, int ldout, int D,
                  const float* g, const float* b, int do_relu) {
        k_fill<<<1, TB, 0, stream>>>(bsum, 0.0f, 2 * P1);   // bsum & bsq contiguous
        const int rows = 128;
        k_bn_stats<<<(N + rows - 1) / rows, 128, 0, stream>>>(in, N, ldin, D, rows, bsum, bsq);
        k_bn_apply<<<((size_t)N * ldout + TB - 1) / TB, TB, 0, stream>>>(
            in, ldin, o, ldout, N, D, bsum, bsq, g, b, 1.0f / (float)N, do_relu);
    };

    // ---- layer 1: relu(gcn_conv(x, W1)) ----
    conv(x, FIN, FIN, conv1_w, FIN, conv1_b, hA);
    float* cur = hA; float* oth = hB;

    // ---- layers 2..4: bn(relu(gcn_conv(h))) ----
    for (int i = 0; i < 3; ++i) {
        conv(cur, P1, P1, convs_w + (size_t)i * D1 * D1, D1, convs_b + (size_t)i * D1, oth);
        bn(oth, P1, oth, P1, D1, bn1_g, bn1_b, /*relu=*/0);
        float* t = cur; cur = oth; oth = t;
    }

    // ---- fc1 (90->80) + bn2 + relu ----
    gemm(cur, P1, P1, fc1_w, D1, D2, fc1_b, h2, P2, P2 / 16);
    bn(h2, P2, oth, P2, D2, bn2_g, bn2_b, /*relu=*/1);

    // ---- fc2 (80->50) + bn3 + relu ----
    gemm(oth, P2, P2, fc2_w, D2, D3, fc2_b, h2, P3, P3 / 16);
    bn(h2, P3, cur, P3, D3, bn3_g, bn3_b, /*relu=*/1);

    // ---- fc3 (50->1) ----
    k_fc3<<<(N + TB - 1) / TB, TB, 0, stream>>>(cur, P3, D3, fc3_w, fc3_b, out, N);
}